// CtdetLoss_41824391529151
// MI455X (gfx1250) — compile-verified
//
#include <hip/hip_runtime.h>
#include <math.h>

#define Bn 16
#define Cn 80
#define Hn 128
#define Wn 128
#define Kn 128
#define HW (Hn * Wn)

typedef float v2f __attribute__((ext_vector_type(2)));
typedef float v4f __attribute__((ext_vector_type(4)));
typedef float v8f __attribute__((ext_vector_type(8)));

// ---------------------------------------------------------------------------
// Kernel 1: Zsum[b,y,x] = sum_c log1p(-p) * p^2   (p = pred_hm[b,c,y,x])
// Bandwidth-bound: streams the full 80 MB pred_hm once. Each thread owns 4
// consecutive pixels -> global_load_b128, 512B per wave per issue, fully
// coalesced; unrolled so multiple b128 loads are in flight to cover latency.
// ---------------------------------------------------------------------------
__global__ void zsum_kernel(const float* __restrict__ pred_hm,
                            float* __restrict__ zsum) {
    int q = blockIdx.x * blockDim.x + threadIdx.x;   // 0 .. B*H*W/4 - 1
    if (q >= (Bn * HW) / 4) return;
    int pix = q * 4;                                 // first of 4 pixels
    int b   = pix / HW;
    int px  = pix - b * HW;
    const v4f* base = (const v4f*)(pred_hm + (size_t)b * Cn * HW + px);
    v4f acc = {0.0f, 0.0f, 0.0f, 0.0f};
#pragma unroll 8
    for (int c = 0; c < Cn; ++c) {
        v4f p = base[(size_t)c * (HW / 4)];
        v4f l;
        l.x = log1pf(-p.x);
        l.y = log1pf(-p.y);
        l.z = log1pf(-p.z);
        l.w = log1pf(-p.w);
        acc += l * p * p;
    }
    *((v4f*)(zsum + pix)) = acc;
}

// ---------------------------------------------------------------------------
// Kernel 2: per-object rectangle sums (replaces the SAT corner lookups).
// One wave32 per object, 8 objects per 256-thread block.
// ---------------------------------------------------------------------------
__global__ void rect_kernel(const float* __restrict__ pred_hm,
                            const float* __restrict__ hm,
                            const float* __restrict__ zsum,
                            const float* __restrict__ wh_t,
                            const float* __restrict__ reg_mask,
                            const int*   __restrict__ cxcy,
                            const int*   __restrict__ ori_wh,
                            const int*   __restrict__ cls_idx,
                            float* __restrict__ contrib) {
    int wave = threadIdx.x >> 5;
    int lane = threadIdx.x & 31;
    int obj  = blockIdx.x * 8 + wave;                // 0 .. B*K-1
    int b = obj / Kn;

    int w  = (int)wh_t[obj * 2 + 0];
    int h  = (int)wh_t[obj * 2 + 1];
    int cx = cxcy[obj * 2 + 0];
    int cy = cxcy[obj * 2 + 1];
    int c  = cls_idx[obj];

    int y0 = max(1, cy - h / 2 - 1);
    int y1 = min(Hn - 1, cy + h / 2 + 1);
    int x0 = max(1, cx - w / 2 - 1);
    int x1 = min(Wn - 1, cx + w / 2 + 1);
    y1 = max(y1, y0);
    x1 = max(x1, x0);
    int rw = x1 - x0;
    int area = rw * (y1 - y0);                       // half-open rect

    const float* ph = pred_hm + ((size_t)b * Cn + c) * HW;
    const float* gh = hm      + ((size_t)b * Cn + c) * HW;
    const float* zb = zsum    + (size_t)b * HW;

    float sF = 0.0f, sZc = 0.0f, np = 0.0f, sZS = 0.0f;
    for (int i = lane; i < area; i += 32) {
        int yy  = y0 + i / rw;
        int xx  = x0 + i % rw;
        int idx = yy * Wn + xx;
        float p = ph[idx];
        float g = gh[idx];
        float z = log1pf(-p) * p * p;                // Z_map element
        sZc += z;
        sZS += zb[idx];
        if (g == 1.0f) {
            float om = 1.0f - p;
            sF += logf(p) * om * om;                 // pos focal
            np += 1.0f;
        } else {                                     // g < 1.0
            float omg = 1.0f - g;
            float w4  = omg * omg;
            w4 *= w4;
            sF += z * w4;                            // neg focal
        }
    }
    // wave32 reduction
    for (int off = 16; off; off >>= 1) {
        sF  += __shfl_down(sF,  off);
        sZc += __shfl_down(sZc, off);
        np  += __shfl_down(np,  off);
        sZS += __shfl_down(sZS, off);
    }
    if (lane == 0) {
        float total    = sF + (sZS - sZc);
        float obj_loss = (np > 0.0f) ? (-total / fmaxf(np, 1.0f)) : (-total);
        float aspect   = (float)w / (float)h;
        float oa = (float)ori_wh[obj * 2 + 0] / (float)ori_wh[obj * 2 + 1];
        bool good = (aspect > 0.5f * oa) && (aspect < 2.0f * oa);
        obj_loss *= good ? 1.0f : 0.5f;
        float valid = reg_mask[obj] * ((w * h > 0) ? 1.0f : 0.0f);
        contrib[obj] = obj_loss * valid;
    }
}

// ---------------------------------------------------------------------------
// Kernel 3: finalization. One wave32. Per-batch sums via V_WMMA_F32_16X16X4_F32
// with B = all-ones (D[m,n] = row-sum of A). 16 batches map onto M=0..15.
// A layout (ISA 7.12.2, 32-bit A 16x4): lanes 0-15 hold M=lane, K={0,1};
// lanes 16-31 hold M=lane-16, K={2,3}. B is all ones -> layout-agnostic.
// ---------------------------------------------------------------------------
__device__ __forceinline__ v8f wmma_rowsum_acc(v2f a, v8f c) {
    v2f ones;
    ones.x = 1.0f;
    ones.y = 1.0f;
    return __builtin_amdgcn_wmma_f32_16x16x4_f32(
        /*neg_a=*/false, a, /*neg_b=*/false, ones,
        /*c_mod=*/(short)0, c, /*reuse_a=*/false, /*reuse_b=*/false);
}

__global__ void final_kernel(const float* __restrict__ contrib,
                             const float* __restrict__ pred_wh,
                             const float* __restrict__ pred_reg,
                             const float* __restrict__ wh_t,
                             const float* __restrict__ reg_t,
                             const float* __restrict__ reg_mask,
                             const int*   __restrict__ ind,
                             float* __restrict__ out) {
    int lane = threadIdx.x;          // 0..31, all active (EXEC all ones)
    int m  = lane & 15;              // batch row this lane feeds
    int kA = (lane >> 4) * 2;        // K slots {0,1} or {2,3}

    v8f cImg = {}; v8f cNob = {}; v8f cWh = {}; v8f cOff = {};

    // img_hm[b] = sum_k contrib ; nobj[b] = sum_k reg_mask  (K=128, 4 per WMMA)
    for (int j0 = 0; j0 < Kn; j0 += 4) {
        int k0 = j0 + kA;
        v2f a, an;
        a.x  = contrib[m * Kn + k0];
        a.y  = contrib[m * Kn + k0 + 1];
        an.x = reg_mask[m * Kn + k0];
        an.y = reg_mask[m * Kn + k0 + 1];
        cImg = wmma_rowsum_acc(a,  cImg);
        cNob = wmma_rowsum_acc(an, cNob);
    }

    // RegL1 numerators: 2*K = 256 |p*m - t*m| terms per batch, 4 per WMMA
    for (int j0 = 0; j0 < 2 * Kn; j0 += 4) {
        v2f aw, ao;
        aw.x = aw.y = ao.x = ao.y = 0.0f;
        for (int t = 0; t < 2; ++t) {
            int j    = j0 + kA + t;
            int k    = j >> 1;
            int comp = j & 1;
            int   id = ind[m * Kn + k];
            float mk = reg_mask[m * Kn + k];
            float pw = pred_wh [((size_t)m * 2 + comp) * HW + id];
            float po = pred_reg[((size_t)m * 2 + comp) * HW + id];
            float tw = wh_t [(m * Kn + k) * 2 + comp];
            float to = reg_t[(m * Kn + k) * 2 + comp];
            float vw = fabsf(pw * mk - tw * mk);
            float vo = fabsf(po * mk - to * mk);
            if (t == 0) { aw.x = vw; ao.x = vo; }
            else        { aw.y = vw; ao.y = vo; }
        }
        cWh  = wmma_rowsum_acc(aw, cWh);
        cOff = wmma_rowsum_acc(ao, cOff);
    }

    // D layout: lane 0 holds batches 0..7 in c[0..7]; lane 16 holds 8..15.
    float pl = 0.0f, phm = 0.0f, pwh = 0.0f, pof = 0.0f;
    if ((lane & 15) == 0) {
        for (int r = 0; r < 8; ++r) {
            float img  = cImg[r];
            float nob  = cNob[r];
            float den  = 2.0f * nob + 1e-4f;            // sum of m2 over (K,2)
            float wh_i = cWh[r]  / den;
            float of_i = cOff[r] / den;
            pl  += 1.0f * img + 0.1f * wh_i + 1.0f * of_i;   // uses UN-normalized img_hm
            phm += (nob > 0.0f) ? (img / fmaxf(nob, 1.0f)) : img;
            pwh += wh_i;
            pof += of_i;
        }
    }
    pl  += __shfl_down(pl,  16);
    phm += __shfl_down(phm, 16);
    pwh += __shfl_down(pwh, 16);
    pof += __shfl_down(pof, 16);
    if (lane == 0) {
        out[0] = pl  * (1.0f / 16.0f);   // final_loss
        out[1] = phm * (1.0f / 16.0f);   // final_hm
        out[2] = pwh * (1.0f / 16.0f);   // final_wh
        out[3] = pof * (1.0f / 16.0f);   // final_off
    }
}

// ---------------------------------------------------------------------------
extern "C" void kernel_launch(void* const* d_in, const int* in_sizes, int n_in,
                              void* d_out, int out_size, void* d_ws, size_t ws_size,
                              hipStream_t stream) {
    const float* pred_hm  = (const float*)d_in[0];
    const float* pred_wh  = (const float*)d_in[1];
    const float* pred_reg = (const float*)d_in[2];
    const float* hm       = (const float*)d_in[3];
    const float* wh_t     = (const float*)d_in[4];
    const float* reg_t    = (const float*)d_in[5];
    const float* reg_mask = (const float*)d_in[6];
    const int*   ind      = (const int*)d_in[7];
    const int*   cxcy     = (const int*)d_in[8];
    const int*   ori_wh   = (const int*)d_in[9];
    const int*   cls_idx  = (const int*)d_in[10];
    float* out = (float*)d_out;

    float* zsum    = (float*)d_ws;            // B*H*W floats (1 MB)
    float* contrib = zsum + (size_t)Bn * HW;  // B*K floats (8 KB)

    zsum_kernel<<<(Bn * HW / 4 + 255) / 256, 256, 0, stream>>>(pred_hm, zsum);
    rect_kernel<<<(Bn * Kn) / 8, 256, 0, stream>>>(pred_hm, hm, zsum, wh_t,
                                                   reg_mask, cxcy, ori_wh,
                                                   cls_idx, contrib);
    final_kernel<<<1, 32, 0, stream>>>(contrib, pred_wh, pred_reg, wh_t,
                                       reg_t, reg_mask, ind, out);
}